// GINNet_79216376808033
// MI455X (gfx1250) — compile-verified
//
#include <hip/hip_runtime.h>
#include <math.h>

// -------- CDNA5 / gfx1250 GINE-GNN: fused WMMA edge-GEMM + L2-resident scatter --------
//
// All GEMMs use V_WMMA_F32_16X16X4_F32 (full fp32, D = A[16x4]*B[4x16] + C).
// VGPR layouts per CDNA5 ISA 7.12.2:
//   A (16x4 f32):  lane l, vgpr v  -> A[l%16][v + 2*(l>=16)]
//   B (4x16 f32):  lane l, vgpr v  -> B[v + 2*(l>=16)][l%16]
//   C/D (16x16):   lane l, vgpr v  -> D[v + 8*(l>=16)][l%16]
//
// Memory policy: edge streams (edge_attr, edge_index) are single-use ->
// nontemporal loads (TH_NT) to keep the 192MB L2 reserved for the 25.6MB
// node tables that the gather + fp32-atomic scatter hammer.
// Node-table accesses use 32-bit byte offsets (tables < 2^31 bytes) so the
// compiler emits saddr+voffset32 (GVS) addressing (confirmed in round-2 asm).
// Gathers are issued ahead of the WMMA chain so their L2 latency overlaps
// the 8-deep WMMA accumulation instead of serializing with the atomics.

typedef __attribute__((ext_vector_type(2))) float v2f;
typedef __attribute__((ext_vector_type(8))) float v8f;

static constexpr int NNODES = 100000;   // multiple of 16
static constexpr int NEDGES = 1600000;  // multiple of 16
static constexpr int NGRAPH = 256;

__device__ __forceinline__ v8f wmma4(v2f a, v2f b, v8f c) {
  // 8 args: (neg_a, A, neg_b, B, c_mod, C, reuse_a, reuse_b)
  return __builtin_amdgcn_wmma_f32_16x16x4_f32(false, a, false, b, (short)0, c,
                                               false, false);
}

__device__ __forceinline__ float ld_off(const float* base, unsigned byteOff) {
  return *(const float*)((const char*)base + byteOff);
}
__device__ __forceinline__ v2f ld2_nt_off(const float* base, unsigned byteOff) {
  return __builtin_nontemporal_load((const v2f*)((const char*)base + byteOff));
}
__device__ __forceinline__ void st_off(float* base, unsigned byteOff, float v) {
  *(float*)((char*)base + byteOff) = v;
}
__device__ __forceinline__ void atomic_add_off(float* base, unsigned byteOff,
                                               float v) {
  unsafeAtomicAdd((float*)((char*)base + byteOff), v);  // global_atomic_add_f32
}

// One wave processes a 16-edge tile:
//   e = edge_attr[tile,0:32] @ W[32,D] + bias   (WMMA, K=32 in 8 chunks of 4)
//   msg = relu(h_in[src] + e); atomicAdd(agg[dst], msg)
template <int D>
__global__ __launch_bounds__(256) void edge_scatter(
    const float* __restrict__ ea,    // [NEDGES, 32]
    const float* __restrict__ W,     // [32, D]
    const float* __restrict__ bias,  // [D]
    const float* __restrict__ hin,   // [NNODES, D]
    const int* __restrict__ src, const int* __restrict__ dst,
    float* __restrict__ agg) {       // [NNODES, D]
  constexpr int NC = D / 16;
  const int lane  = threadIdx.x & 31;
  const int col16 = lane & 15;
  const int hi    = lane >> 4;      // half-wave id
  const int kof   = hi * 2;
  const int wid   = (blockIdx.x * blockDim.x + threadIdx.x) >> 5;
  const int nw    = (gridDim.x * blockDim.x) >> 5;

  // Preload B-matrix (weights) and bias into registers (tiny; L2-hot).
  v2f   breg[NC][8];
  float bv[NC];
#pragma unroll
  for (int nc = 0; nc < NC; ++nc) {
    const int c = nc * 16 + col16;
    bv[nc] = bias[c];
#pragma unroll
    for (int kc = 0; kc < 8; ++kc) {
      const int k0 = kc * 4 + kof;
      breg[nc][kc].x = W[k0 * D + c];
      breg[nc][kc].y = W[(k0 + 1) * D + c];
    }
  }

  const int ntiles = NEDGES / 16;
  for (int t = wid; t < ntiles; t += nw) {
    const int      eb    = t * 16;
    const unsigned arowB = (unsigned)(eb + col16) * 128u;  // row * 32 floats * 4B
    v2f areg[8];
#pragma unroll
    for (int kc = 0; kc < 8; ++kc) {  // streaming edge_attr: TH_NT, 8B pairs
      areg[kc] = ld2_nt_off(ea, arowB + (unsigned)(kc * 16 + kof * 4));
    }
    unsigned sB[8], dB[8];
#pragma unroll
    for (int v = 0; v < 8; ++v) {  // edge index stream: TH_NT
      const int m = eb + v + 8 * hi;
      sB[v] = (unsigned)__builtin_nontemporal_load(src + m) * (unsigned)(D * 4);
      dB[v] = (unsigned)__builtin_nontemporal_load(dst + m) * (unsigned)(D * 4);
    }
#pragma unroll
    for (int nc = 0; nc < NC; ++nc) {
      const unsigned cB = (unsigned)(nc * 16 + col16) * 4u;
      // Issue all 8 gathers first: they clause together and their L2 latency
      // is hidden behind the 8-WMMA accumulation chain below.
      float gv[8];
#pragma unroll
      for (int v = 0; v < 8; ++v) gv[v] = ld_off(hin, sB[v] + cB);
      v8f acc;
#pragma unroll
      for (int v = 0; v < 8; ++v) acc[v] = bv[nc];  // bias per output column
#pragma unroll
      for (int kc = 0; kc < 8; ++kc) acc = wmma4(areg[kc], breg[nc][kc], acc);
#pragma unroll
      for (int v = 0; v < 8; ++v) {
        float msg = gv[v] + acc[v];
        msg = msg > 0.0f ? msg : 0.0f;
        atomic_add_off(agg, dB[v] + cB, msg);  // L2-resident fp32 atomic
      }
    }
  }
}

// h_out = elu( ((1+eps)*h + agg) @ W[DIN,DOUT] + b )
template <int DIN, int DOUT>
__global__ __launch_bounds__(256) void node_update(
    const float* __restrict__ h, const float* __restrict__ agg,
    const float* __restrict__ W, const float* __restrict__ bias,
    const float* __restrict__ epsp, float* __restrict__ hout) {
  constexpr int KC = DIN / 4;
  constexpr int NC = DOUT / 16;
  const int lane  = threadIdx.x & 31;
  const int col16 = lane & 15;
  const int hi    = lane >> 4;
  const int kof   = hi * 2;
  const int wid   = (blockIdx.x * blockDim.x + threadIdx.x) >> 5;
  const int nw    = (gridDim.x * blockDim.x) >> 5;
  const float oe  = 1.0f + epsp[0];

  v2f   breg[NC][KC];
  float bv[NC];
#pragma unroll
  for (int nc = 0; nc < NC; ++nc) {
    const int c = nc * 16 + col16;
    bv[nc] = bias[c];
#pragma unroll
    for (int kc = 0; kc < KC; ++kc) {
      const int k0 = kc * 4 + kof;
      breg[nc][kc].x = W[k0 * DOUT + c];
      breg[nc][kc].y = W[(k0 + 1) * DOUT + c];
    }
  }

  const int ntiles = NNODES / 16;
  for (int t = wid; t < ntiles; t += nw) {
    const int      rb    = t * 16;
    const unsigned arowB = (unsigned)(rb + col16) * (unsigned)(DIN * 4);
    v2f areg[KC];
#pragma unroll
    for (int kc = 0; kc < KC; ++kc) {  // fuse (1+eps)*h + agg into A operand
      const unsigned o = arowB + (unsigned)(kc * 16 + kof * 4);
      const float h0 = ld_off(h, o),      h1 = ld_off(h, o + 4);
      const float a0 = ld_off(agg, o),    a1 = ld_off(agg, o + 4);
      areg[kc].x = oe * h0 + a0;
      areg[kc].y = oe * h1 + a1;
    }
#pragma unroll
    for (int nc = 0; nc < NC; ++nc) {
      v8f acc;
#pragma unroll
      for (int v = 0; v < 8; ++v) acc[v] = bv[nc];
#pragma unroll
      for (int kc = 0; kc < KC; ++kc) acc = wmma4(areg[kc], breg[nc][kc], acc);
      const unsigned cB = (unsigned)(nc * 16 + col16) * 4u;
#pragma unroll
      for (int v = 0; v < 8; ++v) {
        float d = acc[v];
        d = d > 0.0f ? d : expm1f(d);  // ELU(alpha=1)
        st_off(hout,
               (unsigned)(rb + v + 8 * hi) * (unsigned)(DOUT * 4) + cB, d);
      }
    }
  }
}

// Per-graph max-pool over sorted batch ids: one wave per graph, lane = dim.
__global__ __launch_bounds__(32) void pool_max(const float* __restrict__ h,
                                               const int* __restrict__ batch,
                                               float* __restrict__ out) {
  const int g    = blockIdx.x;
  const int lane = threadIdx.x;  // 0..31 = feature dim
  int lo = 0, hi = NNODES;
  while (lo < hi) { int mid = (lo + hi) >> 1; if (batch[mid] < g) lo = mid + 1; else hi = mid; }
  const int s = lo;
  hi = NNODES;
  while (lo < hi) { int mid = (lo + hi) >> 1; if (batch[mid] < g + 1) lo = mid + 1; else hi = mid; }
  const int e = lo;
  float m = -__builtin_inff();  // segment_max identity
  for (int n = s; n < e; ++n)
    m = fmaxf(m, ld_off(h, (unsigned)n * 128u + (unsigned)lane * 4u));
  out[g * 32 + lane] = m;
}

extern "C" void kernel_launch(void* const* d_in, const int* in_sizes, int n_in,
                              void* d_out, int out_size, void* d_ws, size_t ws_size,
                              hipStream_t stream) {
  (void)in_sizes; (void)n_in; (void)out_size; (void)ws_size;
  const float* x     = (const float*)d_in[0];
  const int*   src   = (const int*)d_in[1];          // edge_index[0]
  const int*   dst   = src + NEDGES;                 // edge_index[1]
  const float* ea    = (const float*)d_in[2];
  const int*   batch = (const int*)d_in[3];
  const float* fc1W  = (const float*)d_in[4];  const float* fc1b = (const float*)d_in[5];
  const float* fc2W  = (const float*)d_in[6];  const float* fc2b = (const float*)d_in[7];
  const float* fc3W  = (const float*)d_in[8];  const float* fc3b = (const float*)d_in[9];
  const float* W1    = (const float*)d_in[10]; const float* b1   = (const float*)d_in[11];
  const float* W2    = (const float*)d_in[12]; const float* b2   = (const float*)d_in[13];
  const float* W3    = (const float*)d_in[14]; const float* b3   = (const float*)d_in[15];
  const float* eps1  = (const float*)d_in[16];
  const float* eps2  = (const float*)d_in[17];
  const float* eps3  = (const float*)d_in[18];
  float* out = (float*)d_out;

  float* agg = (float*)d_ws;                 // [NNODES,64] scratch (reused)
  float* h1  = agg + (size_t)NNODES * 64;    // [NNODES,32]
  float* h2  = h1  + (size_t)NNODES * 32;    // [NNODES,64]
  float* h3  = h2  + (size_t)NNODES * 64;    // [NNODES,32]

  const dim3 blk(256);

  // --- layer 1: e1 dim 64, W1: 64->32 ---
  hipMemsetAsync(agg, 0, sizeof(float) * (size_t)NNODES * 64, stream);
  edge_scatter<64><<<1024, blk, 0, stream>>>(ea, fc1W, fc1b, x, src, dst, agg);
  node_update<64, 32><<<512, blk, 0, stream>>>(x, agg, W1, b1, eps1, h1);

  // --- layer 2: e2 dim 32, W2: 32->64 ---
  hipMemsetAsync(agg, 0, sizeof(float) * (size_t)NNODES * 32, stream);
  edge_scatter<32><<<1024, blk, 0, stream>>>(ea, fc2W, fc2b, h1, src, dst, agg);
  node_update<32, 64><<<512, blk, 0, stream>>>(h1, agg, W2, b2, eps2, h2);

  // --- layer 3: e3 dim 64, W3: 64->32 ---
  hipMemsetAsync(agg, 0, sizeof(float) * (size_t)NNODES * 64, stream);
  edge_scatter<64><<<1024, blk, 0, stream>>>(ea, fc3W, fc3b, h2, src, dst, agg);
  node_update<64, 32><<<512, blk, 0, stream>>>(h2, agg, W3, b3, eps3, h3);

  // --- global max pool ---
  pool_max<<<NGRAPH, 32, 0, stream>>>(h3, batch, out);
}